// LearnableTopP_84516366451096
// MI455X (gfx1250) — compile-verified
//
#include <hip/hip_runtime.h>
#include <stdint.h>

// LearnableTopP: per-row descending sort + cumsum + threshold mask.
// B=4, H=16, S=1024  ->  65,536 independent rows of 1024 floats.
#define S_LEN    1024
#define THREADS  256
#define EPT      4            // elements per thread (1024 / 256)
#define NROWS    65536        // B*H*S
#define NWAVES   8            // 256 / wave32

// Wavefront-synchronous LDS fence: DS ops from one wave retire in order
// (CDNA5 DScnt ordering), so for intra-wave exchanges we only need to stop
// the compiler from caching/reordering LDS accesses.
__device__ __forceinline__ void wave_lds_fence() {
    __builtin_amdgcn_wave_barrier();
    asm volatile("" ::: "memory");
}

__global__ __launch_bounds__(THREADS) void
learnable_topp_kernel(const float* __restrict__ atn,
                      const float* __restrict__ threshold,
                      int* __restrict__ out_idx,   // [NROWS, S]
                      int* __restrict__ out_cnt)   // [NROWS]
{
    __shared__ float              stage[S_LEN];    // 4 KB, async-load target
    __shared__ unsigned long long keys[S_LEN];     // 8 KB sort keys
    __shared__ float              wpart[NWAVES];   // cross-wave scan partials
    __shared__ int                cnt;

    const int row  = blockIdx.x;
    const int t    = threadIdx.x;
    const int lane = t & 31;
    const int wave = t >> 5;
    const float* src = atn + (size_t)row * S_LEN;

    if (t == 0) cnt = 0;

    // ---------------------------------------------------------------
    // Stage the row into LDS via the CDNA5 async-to-LDS path.
    // Each lane moves 16B (4 contiguous floats): 256 lanes * 16B = 4 KB.
    // Generic LDS pointer truncated to 32 bits == LDS byte offset
    // (aperture lives entirely in addr[63:32] on gfx1250).
    // ---------------------------------------------------------------
    {
        unsigned lds_off = (unsigned)(uintptr_t)(&stage[t * EPT]);
        const float* g = src + t * EPT;
        asm volatile("global_load_async_to_lds_b128 %0, %1, off"
                     :: "v"(lds_off), "v"(g)
                     : "memory");
        asm volatile("s_wait_asynccnt 0" ::: "memory");
    }
    __syncthreads();

    // ---------------------------------------------------------------
    // Build 64-bit sort keys. Softmax outputs are positive floats, so
    // their IEEE bit patterns order identically to their values.
    //   key = (value_bits << 32) | (S-1 - idx)
    // Descending key sort => descending value, ties broken by
    // ascending index (matches stable jnp.argsort(-x)).
    // ---------------------------------------------------------------
    #pragma unroll
    for (int e = 0; e < EPT; ++e) {
        int i = t * EPT + e;
        unsigned vb = __float_as_uint(stage[i]);
        keys[i] = ((unsigned long long)vb << 32) | (unsigned)(S_LEN - 1 - i);
    }
    __syncthreads();

    // ---------------------------------------------------------------
    // Bitonic sort (descending) on 1024 keys in LDS.
    // Pass stride j: thread t exchanges pair {i, i^j}. For j < 32 the
    // partner thread t^j is in the same wave32 -> wavefront-synchronous,
    // no s_barrier. Only 15 cross-wave passes + one barrier per k-stage
    // (k >= 32) need full workgroup barriers.
    // ---------------------------------------------------------------
    for (int k = 2; k <= S_LEN; k <<= 1) {
        for (int j = k >> 1; j > 0; j >>= 1) {
            #pragma unroll
            for (int e = 0; e < EPT; ++e) {
                int i = t + e * THREADS;
                int p = i ^ j;
                if (p > i) {
                    unsigned long long a = keys[i];
                    unsigned long long b = keys[p];
                    bool desc = ((i & k) == 0);   // final full merge => all descending
                    if (desc ? (a < b) : (a > b)) {
                        keys[i] = b;
                        keys[p] = a;
                    }
                }
            }
            if (j >= 32) __syncthreads();
            else         wave_lds_fence();
        }
        // Before the next stage's first cross-wave pass (k>=64 has j>=32),
        // and after the final stage, flush the wave-sync tail passes.
        if (k >= 32) __syncthreads();
    }

    // ---------------------------------------------------------------
    // Cumulative sum of sorted values:
    //   per-thread serial over its 4 contiguous slots
    // + intra-wave inclusive scan via __shfl_up (wave32, no barriers)
    // + 8 wave partials combined through LDS (one barrier).
    // ---------------------------------------------------------------
    float v[EPT];
    float run = 0.0f;
    const int base = t * EPT;
    #pragma unroll
    for (int e = 0; e < EPT; ++e) {
        v[e] = __uint_as_float((unsigned)(keys[base + e] >> 32));
        run += v[e];
    }

    float s = run;                       // inclusive scan of thread sums in-wave
    #pragma unroll
    for (int d = 1; d < 32; d <<= 1) {
        float n = __shfl_up(s, d, 32);
        if (lane >= d) s += n;
    }
    if (lane == 31) wpart[wave] = s;     // wave totals
    __syncthreads();

    float woff = 0.0f;
    #pragma unroll
    for (int w = 0; w < NWAVES; ++w)
        woff += (w < wave) ? wpart[w] : 0.0f;

    float csum = woff + s - run;         // exclusive prefix for this thread's chunk

    // sigmoid(threshold[row % 16]); tile(sigmoid(thr), S)[j] == thr[j % H]
    const float th  = threshold[row & 15];
    const float sth = 1.0f / (1.0f + expf(-th));

    // ---------------------------------------------------------------
    // Emit padded indices (selected-in-sorted-order else -1) + count.
    // ---------------------------------------------------------------
    int r[EPT];
    int my = 0;
    #pragma unroll
    for (int e = 0; e < EPT; ++e) {
        csum += v[e];                    // inclusive cumsum at sorted position base+e
        int idx = (S_LEN - 1) - (int)(unsigned)(keys[base + e] & 0xFFFFFFFFull);
        bool m = (csum <= sth);
        r[e] = m ? idx : -1;
        my  += m ? 1 : 0;
    }
    int4 outv;
    outv.x = r[0]; outv.y = r[1]; outv.z = r[2]; outv.w = r[3];
    *reinterpret_cast<int4*>(out_idx + (size_t)row * S_LEN + base) = outv;

    atomicAdd(&cnt, my);
    __syncthreads();
    if (t == 0) out_cnt[row] = cnt;
}

extern "C" void kernel_launch(void* const* d_in, const int* in_sizes, int n_in,
                              void* d_out, int out_size, void* d_ws, size_t ws_size,
                              hipStream_t stream)
{
    const float* atn = (const float*)d_in[0];   // [4,16,1024,1024] f32
    const float* thr = (const float*)d_in[1];   // [16] f32
    int* out_idx = (int*)d_out;                          // [4, 16384, 1024] i32
    int* out_cnt = out_idx + (size_t)NROWS * S_LEN;      // [4, 16384]       i32

    learnable_topp_kernel<<<NROWS, THREADS, 0, stream>>>(atn, thr, out_idx, out_cnt);
}